// BusStopGNN_5222680232375
// MI455X (gfx1250) — compile-verified
//
#include <hip/hip_runtime.h>

// ---------------------------------------------------------------------------
// BusStopGNN on MI455X (gfx1250): 2-layer GCN + linear head, fp32 end-to-end.
// GEMMs use V_WMMA_F32_16X16X4_F32 (wave32); W is pair-packed in LDS so each
// B-fragment is a single aligned ds_load_b64; X staging is padded (stride 68)
// to kill LDS bank conflicts on A-fragment loads. Edge aggregation uses
// hardware global_atomic_add_f32 with wave-scalarized edge metadata loads.
// ---------------------------------------------------------------------------

typedef float v2f __attribute__((ext_vector_type(2)));
typedef float v8f __attribute__((ext_vector_type(8)));

#define GNN_N 100000
#define GNN_E 1600000
#define DIM   64
#define XPAD  68   // padded LDS row stride (floats): bank = (4*lane + k) % 64

// ---------------- utility: fill ----------------
__global__ void fill_kernel(float* __restrict__ p, float v, int n) {
    int i = blockIdx.x * blockDim.x + threadIdx.x;
    if (i < n) p[i] = v;
}

// ---------------- degree (dst occurrences; buffer pre-filled with 1.0 = self loop)
__global__ void deg_kernel(const int* __restrict__ dst, float* __restrict__ deg, int E) {
    int e = blockIdx.x * blockDim.x + threadIdx.x;
    if (e < E)
        __hip_atomic_fetch_add(&deg[dst[e]], 1.0f, __ATOMIC_RELAXED, __HIP_MEMORY_SCOPE_AGENT);
}

__global__ void rsqrt_kernel(float* __restrict__ d, int n) {
    int i = blockIdx.x * blockDim.x + threadIdx.x;
    if (i < n) d[i] = rsqrtf(d[i]);
}

__global__ void norm_kernel(const int* __restrict__ src, const int* __restrict__ dst,
                            const float* __restrict__ dinv, float* __restrict__ nrm, int E) {
    int e = blockIdx.x * blockDim.x + threadIdx.x;
    if (e < E) nrm[e] = dinv[src[e]] * dinv[dst[e]];
}

// ---------------- H = X @ W  (X: nTiles*16 x 64, W: 64 x 64 row-major) -------
// One wave computes a 16x64 output stripe with 4 f32 WMMA accumulators.
// A-frag (16x4 f32): lanes 0-15 hold K=k,k+1 for M=lane; lanes 16-31 hold K=k+2,k+3.
// B-frag (4x16 f32): lanes 0-15 hold rows K=k,k+1 at N=lane; lanes 16-31 rows k+2,k+3.
// W in LDS is pair-packed: Ws[(k>>1)*128 + n*2 + (k&1)] so a B-frag is one b64 load.
// C/D (16x16 f32): VGPR r -> M=r (lanes 0-15) / M=r+8 (lanes 16-31), N=lane%16.
__global__ __launch_bounds__(256) void gemm64_wmma(const float* __restrict__ X,
                                                   const float* __restrict__ W,
                                                   float* __restrict__ H, int nTiles) {
    __shared__ float Ws[64 * 64];          // 16 KB, pair-packed
    __shared__ float Xs[8][16 * XPAD];     // 8 waves x 4.25 KB, padded rows

    const int tid  = threadIdx.x;
    const int wave = tid >> 5;
    const int lane = tid & 31;
    const int tile = blockIdx.x * 8 + wave;

    // stage W pair-packed (whole block; coalesced global reads)
#pragma unroll
    for (int i = 0; i < 16; ++i) {
        int g = tid + i * 256;
        int k = g >> 6;
        int n = g & 63;
        Ws[(k >> 1) * 128 + n * 2 + (k & 1)] = W[g];
    }

    // stage this wave's 16x64 X tile into padded LDS rows (coalesced float4)
    if (tile < nTiles) {
        const float4* Xg = (const float4*)(X + (size_t)tile * 16 * 64);
        float* xl = &Xs[wave][0];
#pragma unroll
        for (int i = 0; i < 8; ++i) {
            int idx4 = lane + i * 32;
            float4 v = Xg[idx4];
            int row = idx4 >> 4;
            int col = (idx4 & 15) * 4;
            *(float4*)&xl[row * XPAD + col] = v;
        }
    }
    __syncthreads();
    if (tile >= nTiles) return;

    const float* xs = &Xs[wave][0];
    const int l15 = lane & 15;
    const int hiK = (lane >> 4) * 2;       // 0 for lanes 0-15, 2 for lanes 16-31

    v8f c0 = {}, c1 = {}, c2 = {}, c3 = {};
#pragma unroll
    for (int k = 0; k < 64; k += 4) {
        v2f a = *(const v2f*)(xs + l15 * XPAD + k + hiK);
        const float* wrow = Ws + ((k + hiK) >> 1) * 128 + l15 * 2;
        v2f b0 = *(const v2f*)(wrow +  0);
        v2f b1 = *(const v2f*)(wrow + 32);
        v2f b2 = *(const v2f*)(wrow + 64);
        v2f b3 = *(const v2f*)(wrow + 96);
        c0 = __builtin_amdgcn_wmma_f32_16x16x4_f32(false, a, false, b0, (short)0, c0, false, false);
        c1 = __builtin_amdgcn_wmma_f32_16x16x4_f32(false, a, false, b1, (short)0, c1, false, false);
        c2 = __builtin_amdgcn_wmma_f32_16x16x4_f32(false, a, false, b2, (short)0, c2, false, false);
        c3 = __builtin_amdgcn_wmma_f32_16x16x4_f32(false, a, false, b3, (short)0, c3, false, false);
    }

    const size_t base = (size_t)tile * 16;
    const int rowOff = (lane >> 4) * 8;
#pragma unroll
    for (int r = 0; r < 8; ++r) {
        float* out = H + (base + r + rowOff) * 64 + l15;
        out[0]  = c0[r];
        out[16] = c1[r];
        out[32] = c2[r];
        out[48] = c3[r];
    }
}

// ---------------- edge scatter: agg[dst] += h[src] * norm[e] -----------------
// 64 threads per edge (one dim per lane): edge id is wave-uniform -> force it
// into an SGPR so src/dst/norm become scalar (s_load) reads; the gather is a
// coalesced 256B row read; accumulation is hw global_atomic_add_f32.
__global__ __launch_bounds__(256) void scatter_kernel(const float* __restrict__ h,
                                                      const float* __restrict__ nrm,
                                                      const int* __restrict__ src,
                                                      const int* __restrict__ dst,
                                                      float* __restrict__ agg, int E) {
    int ew = __builtin_amdgcn_readfirstlane((int)threadIdx.x >> 6);  // wave-uniform
    int e  = blockIdx.x * 4 + ew;
    int d  = threadIdx.x & 63;
    if (e >= E) return;
    int   s = src[e];
    int   q = dst[e];
    float w = nrm[e];
    float v = h[(size_t)s * 64 + d] * w;
    __hip_atomic_fetch_add(&agg[(size_t)q * 64 + d], v,
                           __ATOMIC_RELAXED, __HIP_MEMORY_SCOPE_AGENT);
}

// ---------------- out = maybe_relu(agg + dinv^2 * h + b)  (self-loop folded) -
template <bool RELU>
__global__ void finalize_kernel(float* __restrict__ agg, const float* __restrict__ h,
                                const float* __restrict__ dinv, const float* __restrict__ b,
                                int total) {
    int t = blockIdx.x * blockDim.x + threadIdx.x;
    if (t >= total) return;
    int i = __builtin_amdgcn_readfirstlane(t >> 6);   // wave-uniform node id
    int d = t & 63;
    float di = dinv[i];
    float v = agg[t] + di * di * h[t] + b[d];
    if (RELU) v = fmaxf(v, 0.0f);
    agg[t] = v;
}

// ---------------- y[i] = feat[i,:] . Wp + bp  (one wave32 per node) ----------
__global__ __launch_bounds__(256) void proj_kernel(const float* __restrict__ f,
                                                   const float* __restrict__ Wp,
                                                   const float* __restrict__ bp,
                                                   float* __restrict__ y, int N) {
    int node = blockIdx.x * 8 + (threadIdx.x >> 5);
    int lane = threadIdx.x & 31;
    if (node >= N) return;
    const float* fr = f + (size_t)node * 64;
    float s = fr[lane] * Wp[lane] + fr[lane + 32] * Wp[lane + 32];
#pragma unroll
    for (int m = 16; m > 0; m >>= 1) s += __shfl_xor(s, m, 32);
    if (lane == 0) y[node] = s + bp[0];
}

// ---------------------------------------------------------------------------
extern "C" void kernel_launch(void* const* d_in, const int* in_sizes, int n_in,
                              void* d_out, int out_size, void* d_ws, size_t ws_size,
                              hipStream_t stream) {
    (void)in_sizes; (void)n_in; (void)out_size; (void)ws_size;

    const float* x  = (const float*)d_in[0];
    const int*   ei = (const int*)d_in[1];     // (2, E) flat: [src..., dst...]
    const float* W1 = (const float*)d_in[2];
    const float* b1 = (const float*)d_in[3];
    const float* W2 = (const float*)d_in[4];
    const float* b2 = (const float*)d_in[5];
    const float* Wp = (const float*)d_in[6];
    const float* bp = (const float*)d_in[7];
    float* y = (float*)d_out;

    const int N = GNN_N, E = GNN_E;
    const int* src = ei;
    const int* dst = ei + E;

    char* ws = (char*)d_ws;
    size_t off = 0;
    auto take = [&](size_t bytes) {
        void* p = ws + off;
        off += (bytes + 255) & ~(size_t)255;
        return p;
    };
    float* dinv = (float*)take((size_t)N * 4);            // 0.4 MB
    float* nrm  = (float*)take((size_t)E * 4);            // 6.4 MB
    float* bufA = (float*)take((size_t)N * DIM * 4);      // 25.6 MB
    float* bufB = (float*)take((size_t)N * DIM * 4);      // 25.6 MB

    const int total = N * DIM;                            // 6.4M
    const int nTiles = N / 16;                            // 6250
    const int gemmBlocks = (nTiles + 7) / 8;              // 782

    // --- degree / normalization (computed once, reused both layers)
    fill_kernel<<<(N + 255) / 256, 256, 0, stream>>>(dinv, 1.0f, N);   // self loop
    deg_kernel<<<(E + 255) / 256, 256, 0, stream>>>(dst, dinv, E);
    rsqrt_kernel<<<(N + 255) / 256, 256, 0, stream>>>(dinv, N);
    norm_kernel<<<(E + 255) / 256, 256, 0, stream>>>(src, dst, dinv, nrm, E);

    // --- layer 1: h1 = x@W1 ; agg ; relu(agg + dinv^2*h1 + b1)
    gemm64_wmma<<<gemmBlocks, 256, 0, stream>>>(x, W1, bufA, nTiles);
    fill_kernel<<<total / 256, 256, 0, stream>>>(bufB, 0.0f, total);
    scatter_kernel<<<E / 4, 256, 0, stream>>>(bufA, nrm, src, dst, bufB, E);
    finalize_kernel<true><<<total / 256, 256, 0, stream>>>(bufB, bufA, dinv, b1, total);

    // --- layer 2: h2 = out1@W2 ; agg ; relu(agg + dinv^2*h2 + b2)
    gemm64_wmma<<<gemmBlocks, 256, 0, stream>>>(bufB, W2, bufA, nTiles);
    fill_kernel<<<total / 256, 256, 0, stream>>>(bufB, 0.0f, total);
    scatter_kernel<<<E / 4, 256, 0, stream>>>(bufA, nrm, src, dst, bufB, E);
    finalize_kernel<true><<<total / 256, 256, 0, stream>>>(bufB, bufA, dinv, b2, total);

    // --- head: y = feat @ Wp + bp
    proj_kernel<<<(N + 7) / 8, 256, 0, stream>>>(bufB, Wp, bp, y, N);
}